// HeteroGNN_46299747451390
// MI455X (gfx1250) — compile-verified
//
#include <hip/hip_runtime.h>

typedef float v2f __attribute__((ext_vector_type(2)));
typedef float v8f __attribute__((ext_vector_type(8)));

#define NTASK 131072
#define NPE   36864
#define NRT   36864
#define HID   128
#define ASTRIDE 132   // LDS row stride (dwords): bank-conflict-free b64 reads, 16B-aligned stores

// ---------------- elementwise / utility kernels ----------------

__global__ void fill_f32(float* p, float v, int n) {
  int i = blockIdx.x * blockDim.x + threadIdx.x;
  if (i < n) p[i] = v;
}

__global__ void copy_f32(const float* __restrict__ s, float* __restrict__ d, int n) {
  int i = blockIdx.x * blockDim.x + threadIdx.x;
  if (i < n) d[i] = s[i];
}

// x_pe[i] = pe_emb[i % 9]  (jnp.tile of [9,16] embedding)
__global__ void tile_emb(const float* __restrict__ emb, float* __restrict__ out, int nrows) {
  int i = blockIdx.x * blockDim.x + threadIdx.x;
  if (i < nrows * 16) {
    int r = i >> 4, c = i & 15;
    out[i] = emb[(r % 9) * 16 + c];
  }
}

// ---------------- edge aggregation (segment sum / max) ----------------

__global__ void scatter_add(const float* __restrict__ x, const int* __restrict__ src,
                            const int* __restrict__ dst, int E, int K,
                            float* __restrict__ agg) {
  int chunks = K >> 2;
  int t = blockIdx.x * blockDim.x + threadIdx.x;
  if (t >= E * chunks) return;
  int e = t / chunks;
  int c = (t - e * chunks) << 2;
  float4 v = *(const float4*)(x + (size_t)src[e] * K + c);
  float* o = agg + (size_t)dst[e] * K + c;
  atomicAdd(o + 0, v.x); atomicAdd(o + 1, v.y);
  atomicAdd(o + 2, v.z); atomicAdd(o + 3, v.w);
}

__device__ __forceinline__ void atomicMaxF(float* a, float v) {
  // valid for any sign mix when buffer is initialized to -inf
  if (v >= 0.0f) atomicMax((int*)a, __float_as_int(v));
  else           atomicMin((unsigned int*)a, __float_as_uint(v));
}

__global__ void scatter_max(const float* __restrict__ x, const int* __restrict__ src,
                            const int* __restrict__ dst, int E, int K,
                            float* __restrict__ agg) {
  int chunks = K >> 2;
  int t = blockIdx.x * blockDim.x + threadIdx.x;
  if (t >= E * chunks) return;
  int e = t / chunks;
  int c = (t - e * chunks) << 2;
  float4 v = *(const float4*)(x + (size_t)src[e] * K + c);
  float* o = agg + (size_t)dst[e] * K + c;
  atomicMaxF(o + 0, v.x); atomicMaxF(o + 1, v.y);
  atomicMaxF(o + 2, v.z); atomicMaxF(o + 3, v.w);
}

// wsum = a + b (+ c)
__global__ void sum3(const float* __restrict__ a, const float* __restrict__ b,
                     const float* __restrict__ c, float* __restrict__ o, int n) {
  int i = blockIdx.x * blockDim.x + threadIdx.x;
  if (i < n) {
    float v = a[i] + b[i];
    if (c) v += c[i];
    o[i] = v;
  }
}

// ---------------- WMMA f32 GEMM:  C[M,N] (+)= A[M,K] @ W[N,K]^T (+bias) ----------------
// One block owns a 16-row M tile; the 16xK A tile is staged in LDS once and shared by
// all N/16 waves (each wave produces one 16x16 tile). PyG segment-max identity (-inf)
// is rewritten to 0 during staging, fusing the empty-neighborhood fixup into the GEMM.
// Launch: grid = M/16 blocks, block = 2*N threads (N/16 waves).
__global__ void __launch_bounds__(256)
gemm16(const float* __restrict__ A, int lda,
       const float* __restrict__ W, int ldw,
       const float* __restrict__ bias,
       float* __restrict__ C, int N, int K, int accum, int relu) {
  __shared__ float As[16 * ASTRIDE];
  size_t m0 = (size_t)blockIdx.x * 16;

  // cooperative stage of A[m0:m0+16, 0:K] into LDS (float4 chunks)
  int kq = K >> 2;                 // K is a multiple of 4
  for (int t = threadIdx.x; t < 16 * kq; t += blockDim.x) {
    int r = t / kq;
    int c = (t - r * kq) << 2;
    float4 v = *(const float4*)(A + (m0 + r) * (size_t)lda + c);
    v.x = (v.x == -__builtin_inff()) ? 0.f : v.x;
    v.y = (v.y == -__builtin_inff()) ? 0.f : v.y;
    v.z = (v.z == -__builtin_inff()) ? 0.f : v.z;
    v.w = (v.w == -__builtin_inff()) ? 0.f : v.w;
    *(float4*)(As + r * ASTRIDE + c) = v;
  }
  __syncthreads();

  int lane = threadIdx.x & 31;
  int n0   = (threadIdx.x >> 5) * 16;
  int half = lane >> 4;            // selects K-pair within the fragment
  int l16  = lane & 15;            // M row (A) / N col (B)

  const float* as = As + l16 * ASTRIDE + 2 * half;            // ds_load_b64 fragments
  const float* wp = W + (size_t)(n0 + l16) * ldw + 2 * half;  // global b64 fragments
  __builtin_prefetch(wp, 0, 0);    // global_prefetch_b8

  v8f acc = {0.f, 0.f, 0.f, 0.f, 0.f, 0.f, 0.f, 0.f};
#pragma unroll 8
  for (int k = 0; k < K; k += 4) {
    v2f a = *(const v2f*)(as + k);
    v2f b; b.x = wp[k]; b.y = wp[k + 1];
    acc = __builtin_amdgcn_wmma_f32_16x16x4_f32(false, a, false, b,
                                                (short)0, acc, false, false);
  }

  int col = n0 + l16;
  float bv = bias ? bias[col] : 0.0f;
#pragma unroll
  for (int r = 0; r < 8; ++r) {
    size_t row = m0 + r + 8 * half;   // C/D layout: VGPR r -> M = r (+8 for lanes 16..31)
    float* cp = C + row * (size_t)N + col;
    float v = acc[r] + bv;
    if (accum) v += *cp;
    if (relu)  v = fmaxf(v, 0.0f);
    *cp = v;
  }
}

// ---------------- BatchNorm (training stats) helpers, feature dim = 64 ----------------

__global__ void colstats(const float* __restrict__ h, int M,
                         float* __restrict__ sum, float* __restrict__ sumsq) {
  int c = threadIdx.x & 63;
  int r = blockIdx.x * 4 + (threadIdx.x >> 6);
  float s = 0.f, s2 = 0.f;
  for (; r < M; r += gridDim.x * 4) {
    float v = h[(size_t)r * 64 + c];
    s += v; s2 += v * v;
  }
  atomicAdd(&sum[c], s);
  atomicAdd(&sumsq[c], s2);
}

__global__ void bn_relu(float* __restrict__ h, int M,
                        const float* __restrict__ sum, const float* __restrict__ sumsq,
                        const float* __restrict__ w, const float* __restrict__ b) {
  int i = blockIdx.x * blockDim.x + threadIdx.x;
  if (i >= M * 64) return;
  int c = i & 63;
  float mu  = sum[c] / (float)M;
  float var = sumsq[c] / (float)M - mu * mu;
  float v = (h[i] - mu) * rsqrtf(var + 1e-5f) * w[c] + b[c];
  h[i] = fmaxf(v, 0.0f);
}

// final 64 -> 2 linear (tiny; plain VALU dot)
__global__ void head2(const float* __restrict__ h, const float* __restrict__ W3,
                      const float* __restrict__ b3, float* __restrict__ out, int M) {
  int r = blockIdx.x * blockDim.x + threadIdx.x;
  if (r >= M) return;
  const float4* hp = (const float4*)(h + (size_t)r * 64);
  float a0 = 0.f, a1 = 0.f;
#pragma unroll
  for (int k = 0; k < 16; ++k) {
    float4 v  = hp[k];
    float4 w0 = ((const float4*)W3)[k];
    float4 w1 = ((const float4*)(W3 + 64))[k];
    a0 += v.x * w0.x + v.y * w0.y + v.z * w0.z + v.w * w0.w;
    a1 += v.x * w1.x + v.y * w1.y + v.z * w1.z + v.w * w1.w;
  }
  out[(size_t)r * 2 + 0] = a0 + b3[0];
  out[(size_t)r * 2 + 1] = a1 + b3[1];
}

// ---------------- host orchestration ----------------

extern "C" void kernel_launch(void* const* d_in, const int* in_sizes, int n_in,
                              void* d_out_, int out_size, void* d_ws, size_t ws_size,
                              hipStream_t stream) {
  (void)in_sizes; (void)n_in; (void)out_size; (void)ws_size;

  const float* x_task = (const float*)d_in[0];
  const int* eptr[7];
  const int  ecnt[7] = {524288, 524288, 131072, 131072, 98304, 36864, 36864};
  for (int i = 0; i < 7; ++i) eptr[i] = (const int*)d_in[1 + i];
  const float* P = (const float*)d_in[9];   // flattened params pytree (f32)
  float* out = (float*)d_out_;
  float* ws  = (float*)d_ws;

  // ---- workspace carve-up (floats) ----
  size_t off = 0;
  auto alloc = [&](size_t n) { float* p = ws + off; off += n; return p; };
  float* bT[2] = {alloc((size_t)NTASK * HID), alloc((size_t)NTASK * HID)};
  float* bP[2] = {alloc((size_t)NPE * HID),   alloc((size_t)NPE * HID)};
  float* bR[2] = {alloc((size_t)NRT * HID),   alloc((size_t)NRT * HID)};
  float* agg   = alloc((size_t)NTASK * HID);
  float* xpe0  = alloc((size_t)NPE * 16);
  float* xrt0  = alloc((size_t)NRT * 16);
  float* wsum  = alloc(128 * 128);
  float* bsum  = alloc(128);
  float* stats = alloc(256);   // sum1, sumsq1, sum2, sumsq2

  // ---- parameter offsets (jax pytree: dict keys sorted) ----
  // alphabetical etype slots: dep, iface, link, map, rev_dep, rev_iface, rev_map
  const int aSrc[7] = {0, 2, 2, 0, 0, 1, 1};   // 0=task 1=pe 2=router
  const int aDst[7] = {0, 1, 2, 1, 0, 2, 0};
  // edge-input order (d_in[1..7]): dep, rev_dep, map, rev_map, link, iface, rev_iface
  const int e2a[7]  = {0, 4, 3, 6, 2, 1, 5};
  const int eSrc[7] = {0, 0, 0, 1, 2, 2, 1};
  const int eDst[7] = {0, 0, 1, 0, 2, 1, 2};

  size_t Wrel_off[3][7], Wroot_off[3][7], brel_off[3][7];
  size_t po = 0;
  for (int l = 0; l < 3; ++l)
    for (int a = 0; a < 7; ++a) {
      int ds = (l == 0) ? (aSrc[a] == 0 ? 64 : 16) : 128;
      int dd = (l == 0) ? (aDst[a] == 0 ? 64 : 16) : 128;
      Wrel_off[l][a]  = po; po += (size_t)128 * ds;   // Wrel [128, ds]
      Wroot_off[l][a] = po; po += (size_t)128 * dd;   // Wroot [128, dd]
      brel_off[l][a]  = po; po += 128;                // brel [128]
    }
  const float* W1 = P + po;
  const float* W2 = W1 + 64 * 128;
  const float* W3 = W2 + 64 * 64;
  const float* b1 = W3 + 2 * 64;
  const float* b2 = b1 + 64;
  const float* b3 = b2 + 64;
  const float* bn1_b = b3 + 2;
  const float* bn1_w = bn1_b + 64;
  const float* bn2_b = bn1_w + 64;
  const float* bn2_w = bn2_b + 64;
  const float* pe_emb = bn2_w + 64;
  const float* rt_emb = pe_emb + 144;

  auto cdiv = [](long n, int b) { return (unsigned)((n + b - 1) / b); };

  // ---- layer-0 embedding tiles ----
  tile_emb<<<cdiv((long)NPE * 16, 256), 256, 0, stream>>>(pe_emb, xpe0, NPE);
  tile_emb<<<cdiv((long)NRT * 16, 256), 256, 0, stream>>>(rt_emb, xrt0, NRT);

  const float* xs[3] = {x_task, xpe0, xrt0};
  int Ks[3] = {64, 16, 16};
  const int NC[3] = {NTASK, NPE, NRT};

  // ---- 3 hetero-conv layers ----
  for (int l = 0; l < 3; ++l) {
    float* outs[3] = {bT[l & 1], bP[l & 1], bR[l & 1]};
    for (int d = 0; d < 3; ++d) {
      int elist[3], ne = 0;
      for (int i = 0; i < 7; ++i) if (eDst[i] == d) elist[ne++] = i;
      int Kd = Ks[d];

      // root term: out_d = x_d @ (sum Wroot)^T + sum brel
      {
        int a0 = e2a[elist[0]], a1 = e2a[elist[1]];
        const float* wc = (ne == 3) ? P + Wroot_off[l][e2a[elist[2]]] : nullptr;
        const float* bc = (ne == 3) ? P + brel_off[l][e2a[elist[2]]]  : nullptr;
        sum3<<<cdiv(128L * Kd, 256), 256, 0, stream>>>(
            P + Wroot_off[l][a0], P + Wroot_off[l][a1], wc, wsum, 128 * Kd);
        sum3<<<1, 128, 0, stream>>>(
            P + brel_off[l][a0], P + brel_off[l][a1], bc, bsum, 128);
      }
      gemm16<<<NC[d] / 16, 2 * HID, 0, stream>>>(
          xs[d], Kd, wsum, Kd, bsum, outs[d], HID, Kd, /*accum=*/0, /*relu=*/0);

      // edge terms: out_d += aggr(x_src) @ Wrel^T   (relu fused into last)
      for (int j = 0; j < ne; ++j) {
        int i = elist[j];
        int s = eSrc[i], a = e2a[i];
        int Ksrc = Ks[s];
        const int* srcI = eptr[i];
        const int* dstI = eptr[i] + ecnt[i];
        long nAgg = (long)NC[d] * Ksrc;
        long nThr = (long)ecnt[i] * (Ksrc >> 2);
        bool isMax = (i == 0);   // "dep" uses max aggregation
        fill_f32<<<cdiv(nAgg, 256), 256, 0, stream>>>(
            agg, isMax ? -__builtin_inff() : 0.0f, (int)nAgg);
        if (isMax) {
          scatter_max<<<cdiv(nThr, 256), 256, 0, stream>>>(xs[s], srcI, dstI, ecnt[i], Ksrc, agg);
          // -inf -> 0 fixup is fused into the GEMM's LDS staging
        } else {
          scatter_add<<<cdiv(nThr, 256), 256, 0, stream>>>(xs[s], srcI, dstI, ecnt[i], Ksrc, agg);
        }
        gemm16<<<NC[d] / 16, 2 * HID, 0, stream>>>(
            agg, Ksrc, P + Wrel_off[l][a], Ksrc, nullptr, outs[d], HID, Ksrc,
            /*accum=*/1, /*relu=*/(j == ne - 1) ? 1 : 0);
      }
    }
    xs[0] = bT[l & 1]; xs[1] = bP[l & 1]; xs[2] = bR[l & 1];
    Ks[0] = Ks[1] = Ks[2] = HID;
  }

  // ---- MLP head on task nodes (agg buffer reused as scratch) ----
  float* xTf = bT[0];                 // layer 2 wrote index 0
  float* t1 = agg;                    // [NTASK, 64]
  float* t2 = agg + (size_t)NTASK * 64;

  gemm16<<<NTASK / 16, 2 * 64, 0, stream>>>(xTf, 128, W1, 128, b1, t1, 64, 128, 0, 0);
  fill_f32<<<1, 256, 0, stream>>>(stats, 0.0f, 256);
  colstats<<<512, 256, 0, stream>>>(t1, NTASK, stats, stats + 64);
  bn_relu<<<cdiv((long)NTASK * 64, 256), 256, 0, stream>>>(t1, NTASK, stats, stats + 64, bn1_w, bn1_b);

  gemm16<<<NTASK / 16, 2 * 64, 0, stream>>>(t1, 64, W2, 64, b2, t2, 64, 64, 0, 0);
  colstats<<<512, 256, 0, stream>>>(t2, NTASK, stats + 128, stats + 192);
  bn_relu<<<cdiv((long)NTASK * 64, 256), 256, 0, stream>>>(t2, NTASK, stats + 128, stats + 192, bn2_w, bn2_b);

  head2<<<cdiv(NTASK, 256), 256, 0, stream>>>(t2, W3, b3, out, NTASK);

  // ---- pe / router outputs ----
  copy_f32<<<cdiv((long)NPE * HID, 256), 256, 0, stream>>>(
      bP[0], out + (size_t)NTASK * 2, NPE * HID);
  copy_f32<<<cdiv((long)NRT * HID, 256), 256, 0, stream>>>(
      bR[0], out + (size_t)NTASK * 2 + (size_t)NPE * HID, NRT * HID);
}